// ResamplingRenderer_73469710565397
// MI455X (gfx1250) — compile-verified
//
#include <hip/hip_runtime.h>
#include <math.h>

// ---------------------------------------------------------------------------
// MI455X (gfx1250) Fourier-resampling renderer.
// FFT2048 = radix 16*16*8; the radix-16 stages are 16x16 complex matmuls on
// V_WMMA_F32_16X16X4_F32 (16 wmma per stage per wave).  W16 fragments are
// hoisted into registers; stage-2 twiddles live in a 256-entry LDS table;
// stage-3 uses a sincos-free constant 8-point butterfly with a twiddle
// recurrence.  Column FFT kernels pull data with GLOBAL_LOAD_ASYNC_TO_LDS_B64
// (ASYNCcnt) straight into digit-reversed LDS slots.
// ---------------------------------------------------------------------------

typedef __attribute__((ext_vector_type(2))) float v2f;
typedef __attribute__((ext_vector_type(8))) float v8f;

#define PI_F 3.14159265358979323846f

#if defined(__AMDGCN__) && __has_builtin(__builtin_amdgcn_wmma_f32_16x16x4_f32)
#define USE_WMMA_F32 1
#else
#define USE_WMMA_F32 0
#endif

#if defined(__AMDGCN__)
#define USE_ASYNC_LDS 1
#else
#define USE_ASYNC_LDS 0
#endif

__device__ __forceinline__ float2 cmul(float2 a, float2 b) {
  return make_float2(a.x * b.x - a.y * b.y, a.x * b.y + a.y * b.x);
}
__device__ __forceinline__ float2 cadd(float2 a, float2 b) {
  return make_float2(a.x + b.x, a.y + b.y);
}
__device__ __forceinline__ float2 csub(float2 a, float2 b) {
  return make_float2(a.x - b.x, a.y - b.y);
}

// ---------------------------------------------------------------------------
// W16 DFT-matrix WMMA fragments, built once per kernel (8 sincos total).
//   A (16x4):  lane l, vgpr v -> M=l%16, K=v+2*(l/16)
// ---------------------------------------------------------------------------
struct W16Frag {
  v2f cr[4];   // cos
  v2f ci[4];   // sin
  v2f cin[4];  // -sin (f32 WMMA has no A-negate modifier; NEG is CNeg only)
};

__device__ __forceinline__ W16Frag make_w16(float sgn) {
  const int lane = (int)(threadIdx.x & 31u);
  const int ln = lane & 15;
  const int lh = lane >> 4;
  W16Frag f;
#pragma unroll
  for (int kk = 0; kk < 4; ++kk) {
#pragma unroll
    for (int v = 0; v < 2; ++v) {
      int k = kk * 4 + v + 2 * lh;
      float th = sgn * 2.0f * PI_F * (float)(ln * k) * (1.0f / 16.0f);
      float s, c;
      __sincosf(th, &s, &c);
      f.cr[kk][v] = c;
      f.ci[kk][v] = s;
      f.cin[kk][v] = -s;
    }
  }
  return f;
}

// ---------------------------------------------------------------------------
// One 16x16 complex DFT (Y = W16 * X) per wave, 16 independent columns.
// B fetched via loadB(k,n), D written via store(m,n).
//   B (4x16):  lane l, vgpr v -> N=l%16, K=v+2*(l/16)
//   C/D:       lane l, vgpr r -> M=r+8*(l/16), N=l%16
// ---------------------------------------------------------------------------
template <typename LoadB, typename Store>
__device__ __forceinline__ void dft16_wmma(const W16Frag& w, LoadB loadB, Store store) {
  const int lane = (int)(threadIdx.x & 31u);
  const int ln = lane & 15;
  const int lh = lane >> 4;
  v8f yr = {};
  v8f yi = {};
#pragma unroll
  for (int kk = 0; kk < 4; ++kk) {
    v2f br, bi;
#pragma unroll
    for (int v = 0; v < 2; ++v) {
      int k = kk * 4 + v + 2 * lh;
      float xr, xi;
      loadB(k, ln, xr, xi);
      br[v] = xr;
      bi[v] = xi;
    }
#if USE_WMMA_F32
    // Yr = Wr*Xr - Wi*Xi ; Yi = Wr*Xi + Wi*Xr
    yr = __builtin_amdgcn_wmma_f32_16x16x4_f32(false, w.cr[kk], false, br, (short)0, yr, false, false);
    yr = __builtin_amdgcn_wmma_f32_16x16x4_f32(false, w.cin[kk], false, bi, (short)0, yr, false, false);
    yi = __builtin_amdgcn_wmma_f32_16x16x4_f32(false, w.cr[kk], false, bi, (short)0, yi, false, false);
    yi = __builtin_amdgcn_wmma_f32_16x16x4_f32(false, w.ci[kk], false, br, (short)0, yi, false, false);
#else
#pragma unroll
    for (int r = 0; r < 8; ++r) {  // host-pass placeholder only
      yr[r] += w.cr[kk][0] * br[0] + w.cin[kk][0] * bi[0];
      yi[r] += w.cr[kk][0] * bi[0] + w.ci[kk][0] * br[0];
    }
#endif
  }
#pragma unroll
  for (int r = 0; r < 8; ++r) store(r + 8 * lh, ln, yr[r], yi[r]);
}

// Digit-reversal for mixed radix [16,16,8]: n = n0 + 16*n1 + 256*n2
// -> addr = 128*n0 + 8*n1 + n2
__device__ __forceinline__ int drev2048(int n) {
  return ((n & 15) << 7) | (((n >> 4) & 15) << 3) | (n >> 8);
}

// Constant-coefficient 8-point DFT (no transcendentals): even/odd DFT4 + W8.
__device__ __forceinline__ void dft4(float2 a, float2 b, float2 c, float2 d,
                                     float2* F, float sgn) {
  float2 e0 = cadd(a, c), e1 = csub(a, c);
  float2 o0 = cadd(b, d), o1 = csub(b, d);
  float2 iso1 = make_float2(-sgn * o1.y, sgn * o1.x);  // (i*sgn)*o1
  F[0] = cadd(e0, o0);
  F[2] = csub(e0, o0);
  F[1] = cadd(e1, iso1);
  F[3] = csub(e1, iso1);
}

// In-LDS 2048-point complex FFT, 256 threads (8 waves).  sgn=-1 fwd, +1 inv.
__device__ void fft2048_lds(float2* lc, float2* tw2, float sgn, const W16Frag& w) {
  const int t = (int)threadIdx.x;  // 0..255
  const int wv = t >> 5;           // wave id 0..7

  // Stage-2 twiddle table W_256^{jk}, tw2[j*16+k]; 1 sincos per thread.
  {
    int j = t >> 4, k = t & 15;
    float a = sgn * 2.0f * PI_F * (float)(j * k) * (1.0f / 256.0f);
    float s, c;
    __sincosf(a, &s, &c);
    tw2[t] = make_float2(c, s);
  }
  __syncthreads();

  // ---- Stage 1: 128 DFT-16 on contiguous chunks; wave wv -> chunks 16wv..16wv+15
  dft16_wmma(
      w,
      [&](int k, int n, float& xr, float& xi) {
        float2 v = lc[(16 * wv + n) * 16 + k];
        xr = v.x;
        xi = v.y;
      },
      [&](int m, int n, float vr, float vi) { lc[(16 * wv + n) * 16 + m] = make_float2(vr, vi); });
  __syncthreads();

  // ---- Stage 2: blocks of 256; wave wv = block, twiddle from LDS table
  dft16_wmma(
      w,
      [&](int k, int j, float& xr, float& xi) {
        float2 v = lc[wv * 256 + j + 16 * k];
        float2 r = cmul(v, tw2[j * 16 + k]);
        xr = r.x;
        xi = r.y;
      },
      [&](int m, int j, float vr, float vi) { lc[wv * 256 + j + 16 * m] = make_float2(vr, vi); });
  __syncthreads();

  // ---- Stage 3: radix-8 across stride 256, one j per thread.
  {
    int j = t;
    float2 x[8];
    float a1 = sgn * 2.0f * PI_F * (float)j * (1.0f / 2048.0f);
    float s1, c1;
    __sincosf(a1, &s1, &c1);
    float2 wj = make_float2(c1, s1);
    float2 tw = make_float2(1.0f, 0.0f);
#pragma unroll
    for (int k = 0; k < 8; ++k) {  // twiddle recurrence: tw = wj^k
      x[k] = cmul(lc[j + 256 * k], tw);
      tw = cmul(tw, wj);
    }
    __syncthreads();
    float2 E[4], O[4];
    dft4(x[0], x[2], x[4], x[6], E, sgn);
    dft4(x[1], x[3], x[5], x[7], O, sgn);
    const float r2 = 0.70710678118654752f;
    const float2 W8[4] = {make_float2(1.0f, 0.0f), make_float2(r2, sgn * r2),
                          make_float2(0.0f, sgn), make_float2(-r2, sgn * r2)};
#pragma unroll
    for (int m = 0; m < 4; ++m) {
      float2 t8 = cmul(W8[m], O[m]);
      lc[m * 256 + j] = cadd(E[m], t8);
      lc[(m + 4) * 256 + j] = csub(E[m], t8);
    }
  }
  __syncthreads();
}

// Async global->LDS b64 copy of one float2 into a digit-reversed LDS slot.
__device__ __forceinline__ void async_ld_f2(float2* ldsDst, const float2* g) {
#if USE_ASYNC_LDS
  unsigned lds = (unsigned)(uintptr_t)ldsDst;
  unsigned long long ga = (unsigned long long)(uintptr_t)g;
  asm volatile("global_load_async_to_lds_b64 %0, %1, off" ::"v"(lds), "v"(ga) : "memory");
#else
  *ldsDst = *g;
#endif
}
__device__ __forceinline__ void async_wait() {
#if USE_ASYNC_LDS
  asm volatile("s_wait_asynccnt 0x0" ::: "memory");
#endif
}

// ---------------------------------------------------------------------------
// K2: fused centered-pad + ifftshift + row FFT (real rows), keep cols 0..1024
// grid (2048, 8)
// ---------------------------------------------------------------------------
__global__ void __launch_bounds__(256) k_fft_rows(const float* __restrict__ model,
                                                  float2* __restrict__ KA) {
  __shared__ float2 lc[2048];
  __shared__ float2 tw2[256];
  int c = (int)blockIdx.y;
  int y = (int)blockIdx.x;
  int py = (y + 1024) & 2047;  // undo ifftshift
  int sy = py - 768;           // pad start (2048-512+1)/2 = 768
  bool rowLive = (sy >= 0) && (sy < 512);
  for (int n = (int)threadIdx.x; n < 2048; n += 256) {
    float v = 0.0f;
    if (rowLive) {
      int px = (n + 1024) & 2047;
      int sx = px - 768;
      if (sx >= 0 && sx < 512) v = model[((size_t)c * 512 + sy) * 512 + sx];
    }
    lc[drev2048(n)] = make_float2(v, 0.0f);
  }
  W16Frag wf = make_w16(-1.0f);
  fft2048_lds(lc, tw2, -1.0f, wf);
  for (int j = (int)threadIdx.x; j < 1025; j += 256)
    KA[((size_t)c * 2048 + y) * 1025 + j] = lc[j];
}

// ---------------------------------------------------------------------------
// K3: column FFT (complex), grid (1025, 8); async global->LDS gather.
// ---------------------------------------------------------------------------
__global__ void __launch_bounds__(256) k_fft_cols(const float2* __restrict__ KA,
                                                  float2* __restrict__ KB) {
  __shared__ float2 lc[2048];
  __shared__ float2 tw2[256];
  int c = (int)blockIdx.y;
  int j = (int)blockIdx.x;
  const float2* col = KA + (size_t)c * 2048 * 1025 + j;
  for (int n = (int)threadIdx.x; n < 2048; n += 256)
    async_ld_f2(&lc[drev2048(n)], col + (size_t)n * 1025);
  async_wait();
  W16Frag wf = make_w16(-1.0f);
  fft2048_lds(lc, tw2, -1.0f, wf);
  for (int n = (int)threadIdx.x; n < 2048; n += 256)
    KB[((size_t)c * 2048 + n) * 1025 + j] = lc[n];
}

// ---------------------------------------------------------------------------
// K4: tiny direct half-plane DFT of a PSF (np x np taps -> nf x (nf/2+1)).
// ---------------------------------------------------------------------------
__global__ void k_dft_psf(const float* __restrict__ psf, int np, int nf,
                          float2* __restrict__ out) {
  int hc = nf / 2 + 1;
  int tot = nf * hc;
  int idx = (int)(blockIdx.x * blockDim.x + threadIdx.x);
  if (idx >= tot) return;
  int i = idx / hc, j = idx % hc;
  float ky = (float)(i - nf / 2);
  float kx = (float)j;
  float cc = 0.5f * (float)(np - 1);
  float w = -2.0f * PI_F / (float)nf;
  float ar = 0.0f, ai = 0.0f;
  for (int y = 0; y < np; ++y) {
    for (int x = 0; x < np; ++x) {
      float ph = w * (kx * ((float)x - cc) + ky * ((float)y - cc));
      float s, c;
      __sincosf(ph, &s, &c);
      float p = psf[y * np + x];
      ar += p * c;
      ai += p * s;
    }
  }
  out[idx] = make_float2(ar, ai);
}

// Bilinear gather, map_coordinates(order=1, mode=constant, cval=0)
__device__ __forceinline__ float2 bilin(const float2* __restrict__ g, int H, int W,
                                        float r, float cpos) {
  float fr = floorf(r), fc = floorf(cpos);
  int r0 = (int)fr, c0 = (int)fc;
  float wr = r - fr, wc = cpos - fc;
  float2 acc = make_float2(0.0f, 0.0f);
#pragma unroll
  for (int dr = 0; dr < 2; ++dr) {
#pragma unroll
    for (int dc = 0; dc < 2; ++dc) {
      int rr = r0 + dr, cc = c0 + dc;
      float wgt = (dr ? wr : 1.0f - wr) * (dc ? wc : 1.0f - wc);
      if (rr >= 0 && rr < H && cc >= 0 && cc < W) {
        float2 v = g[(size_t)rr * W + cc];
        acc.x += wgt * v.x;
        acc.y += wgt * v.y;
      }
    }
  }
  return acc;
}

// ---------------------------------------------------------------------------
// K6: warp model k-space onto the odd T=2049 grid, shift phase + PSF ratio.
// One thread per (i,j), loops the 8 channels (PSF math amortized).
// ---------------------------------------------------------------------------
__global__ void __launch_bounds__(256) k_resample(const float2* __restrict__ KB,
                                                  const float2* __restrict__ KPM,
                                                  const float2* __restrict__ KPO,
                                                  float2* __restrict__ KT) {
  int idx = (int)(blockIdx.x * 256u + threadIdx.x);
  if (idx >= 2049 * 1025) return;
  int i = idx / 1025, j = idx % 1025;
  const float dk = 2.0f * PI_F / 2048.0f;
  float ky = (float)(i - 1024) * dk;
  float kx = (float)j * dk;
  const float J00 = 0.68944f, J01 = -0.121554f, J10 = 0.121554f, J11 = 0.68944f;
  float kxm = J00 * kx + J10 * ky;
  float kym = J01 * kx + J11 * ky;
  float sgn = (kxm < 0.0f) ? -1.0f : 1.0f;

  float scm = 64.0f / (2.0f * PI_F);
  float2 pm = bilin(KPM, 64, 33, sgn * kym * scm + 32.0f, sgn * kxm * scm);
  if (sgn < 0.0f) pm.y = -pm.y;
  float sco = 128.0f / (2.0f * PI_F);
  float2 po = bilin(KPO, 128, 65, ky * sco + 64.0f, kx * sco);

  float den = pm.x * pm.x + pm.y * pm.y + 1e-30f;
  float qx = (po.x * pm.x + po.y * pm.y) / den;
  float qy = (po.y * pm.x - po.x * pm.y) / den;
  float ph = -(kx * 1.3f + ky * (-2.7f));
  float sph, cph;
  __sincosf(ph, &sph, &cph);
  float Fx = qx * cph - qy * sph;
  float Fy = qx * sph + qy * cph;

  float sc = 2048.0f / (2.0f * PI_F);
  float rows = sgn * kym * sc + 1024.0f;
  float cols = sgn * kxm * sc;
  int pr = (int)fmaxf(0.0f, fminf(2047.0f, rows));
  int pc = (int)fmaxf(0.0f, fminf(1024.0f, cols));
  for (int c = 0; c < 8; ++c) {
    const float2* src = KB + (size_t)c * 2048 * 1025;
    __builtin_prefetch(src + (size_t)pr * 1025 + pc, 0, 0);  // global_prefetch_b8
    float2 v = bilin(src, 2048, 1025, rows, cols);
    if (sgn < 0.0f) v.y = -v.y;
    float2 o;
    o.x = v.x * Fx - v.y * Fy;
    o.y = v.x * Fy + v.y * Fx;
    KT[((size_t)c * 2049 + i) * 1025 + j] = o;
  }
}

// ---------------------------------------------------------------------------
// K7: hermitian wrap + drop last row + ifftshift rows (pure gather).
// ---------------------------------------------------------------------------
__global__ void k_wrap_shift(const float2* __restrict__ KT, float2* __restrict__ KW) {
  size_t idx = (size_t)blockIdx.x * 256u + threadIdx.x;
  if (idx >= (size_t)8 * 2048 * 1025) return;
  int j = (int)(idx % 1025);
  int r = (int)((idx / 1025) % 2048);
  int c = (int)(idx / ((size_t)1025 * 2048));
  int i = (r + 1024) & 2047;  // ifftshift along rows
  const float2* A = KT + (size_t)c * 2049 * 1025;
  float2 v = A[(size_t)i * 1025 + j];
  if (j == 1024) {  // edge column aliases its hermitian mirror
    float2 m = A[(size_t)(2048 - i) * 1025 + 1024];
    v.x += m.x;
    v.y -= m.y;
  }
  if (i == 0) {  // row ky=+M folds onto ky=-M (after its own column fold)
    float2 e = A[(size_t)2048 * 1025 + j];
    if (j == 1024) {
      float2 m2 = A[1024];
      e.x += m2.x;
      e.y -= m2.y;
    }
    v.x += e.x;
    v.y += e.y;
  }
  KW[idx] = v;
}

// ---------------------------------------------------------------------------
// K8: inverse column FFT (scale 1/2048), grid (1025, 8); async gather.
// ---------------------------------------------------------------------------
__global__ void __launch_bounds__(256) k_ifft_cols(const float2* __restrict__ KW,
                                                   float2* __restrict__ KC) {
  __shared__ float2 lc[2048];
  __shared__ float2 tw2[256];
  int c = (int)blockIdx.y;
  int j = (int)blockIdx.x;
  const float2* col = KW + (size_t)c * 2048 * 1025 + j;
  for (int n = (int)threadIdx.x; n < 2048; n += 256)
    async_ld_f2(&lc[drev2048(n)], col + (size_t)n * 1025);
  async_wait();
  W16Frag wf = make_w16(1.0f);
  fft2048_lds(lc, tw2, 1.0f, wf);
  const float s = 1.0f / 2048.0f;
  for (int n = (int)threadIdx.x; n < 2048; n += 256) {
    float2 v = lc[n];
    KC[((size_t)c * 2048 + n) * 1025 + j] = make_float2(v.x * s, v.y * s);
  }
}

// ---------------------------------------------------------------------------
// K9: inverse row FFT with hermitian reconstruction, fused fftshift + trim.
// grid (1024, 8), writes d_out directly.
// ---------------------------------------------------------------------------
__global__ void __launch_bounds__(256) k_ifft_rows(const float2* __restrict__ KC,
                                                   float* __restrict__ out) {
  __shared__ float2 lc[2048];
  __shared__ float2 tw2[256];
  int c = (int)blockIdx.y;
  int oy = (int)blockIdx.x;    // 0..1023
  int y = (1536 + oy) & 2047;  // fftshift + center trim
  const float2* R = KC + (size_t)c * 2048 * 1025;
  for (int n = (int)threadIdx.x; n < 2048; n += 256) {
    float2 v;
    if (n <= 1024) {
      v = R[(size_t)y * 1025 + n];
    } else {
      float2 h = R[(size_t)y * 1025 + (2048 - n)];
      v = make_float2(h.x, -h.y);  // per-row hermitian (holds after y-inverse)
    }
    lc[drev2048(n)] = v;
  }
  W16Frag wf = make_w16(1.0f);
  fft2048_lds(lc, tw2, 1.0f, wf);
  const float s = 1.0f / 2048.0f;
  for (int ox = (int)threadIdx.x; ox < 1024; ox += 256) {
    int x = (1536 + ox) & 2047;
    out[((size_t)c * 1024 + oy) * 1024 + ox] = lc[x].x * s;
  }
}

// ---------------------------------------------------------------------------
extern "C" void kernel_launch(void* const* d_in, const int* in_sizes, int n_in,
                              void* d_out, int out_size, void* d_ws, size_t ws_size,
                              hipStream_t stream) {
  (void)in_sizes; (void)n_in; (void)out_size; (void)ws_size;
  const float* model = (const float*)d_in[0];  // (8,512,512)
  const float* psf_m = (const float*)d_in[1];  // (15,15)
  const float* psf_o = (const float*)d_in[2];  // (25,25)

  // Workspace: two big ping-pong regions + tiny PSF k-grids.
  //   RA: max(KA, KT, KC) = 8*2049*1025 complex
  //   RB: max(KB, KW)     = 8*2048*1025 complex
  float* ws = (float*)d_ws;
  const size_t szA = (size_t)8 * 2049 * 1025 * 2;
  const size_t szB = (size_t)8 * 2048 * 1025 * 2;
  float2* RA = (float2*)ws;
  float2* RB = (float2*)(ws + szA);
  float2* KPM = (float2*)(ws + szA + szB);
  float2* KPO = KPM + 64 * 33;

  dim3 gRows(2048, 8), gCols(1025, 8), gOut(1024, 8);

  // forward: pad + row FFT -> RA, col FFT -> RB
  k_fft_rows<<<gRows, 256, 0, stream>>>(model, RA);
  k_fft_cols<<<gCols, 256, 0, stream>>>(RA, RB);

  // tiny PSF half-plane DFTs
  k_dft_psf<<<(64 * 33 + 255) / 256, 256, 0, stream>>>(psf_m, 15, 64, KPM);
  k_dft_psf<<<(128 * 65 + 255) / 256, 256, 0, stream>>>(psf_o, 25, 128, KPO);

  // warp + shift + PSF ratio -> RA (as 8 x 2049 x 1025)
  int np = 2049 * 1025;
  k_resample<<<(np + 255) / 256, 256, 0, stream>>>(RB, KPM, KPO, RA);

  // hermitian wrap + ifftshift rows -> RB (as 8 x 2048 x 1025)
  size_t nw = (size_t)8 * 2048 * 1025;
  k_wrap_shift<<<(unsigned)((nw + 255) / 256), 256, 0, stream>>>(RA, RB);

  // inverse col FFT -> RA, inverse row FFT + shift + trim -> d_out
  k_ifft_cols<<<gCols, 256, 0, stream>>>(RB, RA);
  k_ifft_rows<<<gOut, 256, 0, stream>>>(RA, (float*)d_out);
}